// RationaleSelectorModel_66941360275917
// MI455X (gfx1250) — compile-verified
//
#include <hip/hip_runtime.h>
#include <math.h>

#define B_    128
#define T_    2048
#define D_    128
#define RHO   0.2f
#define NEGI  -1.0e9f
#define CHUNK 32            // d-elements per TDM chunk
#define SPAD  36            // staged row stride in floats (32 data + 4 pad), bank-safe

typedef __attribute__((ext_vector_type(2))) float        v2f;
typedef __attribute__((ext_vector_type(8))) float        v8f;
typedef __attribute__((ext_vector_type(4))) unsigned int v4u;
typedef __attribute__((ext_vector_type(4))) int          v4i;
typedef __attribute__((ext_vector_type(8))) int          v8i;

__device__ __forceinline__ unsigned rfl(unsigned v) {
  return (unsigned)__builtin_amdgcn_readfirstlane((int)v);
}

// TDM: DMA a 16(t) x 32(d) f32 tile, row stride 128 floats in memory, into LDS
// with 4-DWORD padding after each 32-DWORD row (LDS row stride = 36 floats).
__device__ __forceinline__ void tdm_load_tile(unsigned lds_byte,
                                              unsigned long long gaddr) {
  v4u g0;
  g0[0] = rfl(1u);                                               // count=1
  g0[1] = rfl(lds_byte);                                         // lds_addr
  g0[2] = rfl((unsigned)(gaddr & 0xffffffffull));                // global_addr lo
  g0[3] = rfl(((unsigned)(gaddr >> 32) & 0x01ffffffu) | (2u << 30)); // hi | type=2

  v8i g1;
  // data_size=2(4B)<<16 | pad_enable<<20 | pad_interval=4(32dw)<<22 | pad_amount=3(4dw)<<25
  g1[0] = (int)rfl((2u << 16) | (1u << 20) | (4u << 22) | (3u << 25));
  g1[1] = (int)rfl(((unsigned)CHUNK & 0xffffu) << 16);           // tensor_dim0 lo16
  g1[2] = (int)rfl(16u << 16);                                   // tensor_dim1 lo16 = 16
  g1[3] = (int)rfl((unsigned)CHUNK << 16);                       // tile_dim0 = 32
  g1[4] = (int)rfl(16u);                                         // tile_dim1=16, tile_dim2=0
  g1[5] = (int)rfl((unsigned)D_);                                // tensor_dim0_stride = 128
  g1[6] = 0;
  g1[7] = 0;

  v4i gz4 = {0, 0, 0, 0};
  v8i gz8 = {0, 0, 0, 0, 0, 0, 0, 0};
  __builtin_amdgcn_tensor_load_to_lds(g0, g1, gz4, gz4, gz8, 0);
}

__device__ __forceinline__ float block_reduce_f(float v, int is_sum, float* s_tmp,
                                                float* s_res, int tid) {
  #pragma unroll
  for (int o = 16; o > 0; o >>= 1) {
    float other = __shfl_xor(v, o, 32);
    v = is_sum ? (v + other) : fmaxf(v, other);
  }
  if ((tid & 31) == 0) s_tmp[tid >> 5] = v;
  __syncthreads();
  if (tid == 0) {
    float r = s_tmp[0];
    #pragma unroll
    for (int i = 1; i < 8; ++i) r = is_sum ? (r + s_tmp[i]) : fmaxf(r, s_tmp[i]);
    *s_res = r;
  }
  __syncthreads();
  return *s_res;
}

__device__ __forceinline__ int block_reduce_i(int v, int* s_tmp, int* s_res, int tid) {
  #pragma unroll
  for (int o = 16; o > 0; o >>= 1) v += __shfl_xor(v, o, 32);
  if ((tid & 31) == 0) s_tmp[tid >> 5] = v;
  __syncthreads();
  if (tid == 0) {
    int r = 0;
    #pragma unroll
    for (int i = 0; i < 8; ++i) r += s_tmp[i];
    *s_res = r;
  }
  __syncthreads();
  return *s_res;
}

__launch_bounds__(256)
__global__ void rationale_selector_kernel(const float* __restrict__ emb,
                                          const float* __restrict__ attn,
                                          const float* __restrict__ W,
                                          const float* __restrict__ bptr,
                                          float* __restrict__ out) {
  __shared__ float s_z[T_];                     // scores, then z
  __shared__ float s_stage[8][2][16 * SPAD];    // per-wave double-buffered TDM tiles
  __shared__ float s_w[D_];
  __shared__ float s_ftmp[8];
  __shared__ float s_fres;
  __shared__ int   s_itmp[8];
  __shared__ int   s_ires;

  const int tid  = threadIdx.x;
  const int lane = tid & 31;
  const int wv   = tid >> 5;
  const int bId  = blockIdx.x;

  const float  bias = bptr[0];
  const size_t rowA = (size_t)bId * T_;        // attn / output row base
  const size_t rowE = (size_t)bId * T_ * D_;   // embeddings row base

  if (tid < D_) s_w[tid] = W[tid];
  __syncthreads();

  // ---------------- Phase 1: scores via TDM staging + WMMA f32 16x16x4 --------
  // A(16x4) = W broadcast over M; B(4x16) = emb tile (K=d, N=t).
  const int K0  = (lane >> 4) << 1;   // 0 for lanes 0-15, 2 for lanes 16-31
  const int r16 = lane & 15;

  const unsigned stage_lds0 = (unsigned)(uintptr_t)(&s_stage[wv][0][0]);
  const unsigned stage_lds1 = (unsigned)(uintptr_t)(&s_stage[wv][1][0]);

  for (int j = 0; j < 16; ++j) {
    const int t0 = (wv + 8 * j) * 16;
    v8f c = {};

    const unsigned long long gbase =
        (unsigned long long)(uintptr_t)(emb + rowE + (size_t)t0 * D_);

    tdm_load_tile(stage_lds0, gbase);            // chunk 0 -> buf 0

    #pragma unroll
    for (int ch = 0; ch < 4; ++ch) {
      if (ch < 3) {
        const unsigned dst = ((ch + 1) & 1) ? stage_lds1 : stage_lds0;
        tdm_load_tile(dst, gbase + (unsigned long long)((ch + 1) * CHUNK) * 4ull);
        __builtin_amdgcn_s_wait_tensorcnt(1);
      } else {
        __builtin_amdgcn_s_wait_tensorcnt(0);
      }
      asm volatile("" ::: "memory");   // keep LDS reads below the wait

      const float* buf = &s_stage[wv][ch & 1][0];
      #pragma unroll
      for (int k = 0; k < 8; ++k) {
        const int dl = k * 4 + K0;
        v2f a, bm;
        a.x  = s_w[ch * CHUNK + dl];
        a.y  = s_w[ch * CHUNK + dl + 1];
        bm.x = buf[r16 * SPAD + dl];
        bm.y = buf[r16 * SPAD + dl + 1];
        c = __builtin_amdgcn_wmma_f32_16x16x4_f32(false, a, false, bm,
                                                  (short)0, c, false, false);
      }
    }

    if (lane < 16) {
      const int   t   = t0 + lane;
      const float att = attn[rowA + t];
      float sv = att * c[0] + bias;
      if (att == 0.0f) sv = NEGI;
      s_z[t] = sv;
    }
  }
  __syncthreads();

  // ---------------- Phase 2: softmax, K, z ----------------
  float zreg[8];
  float smax = -3.0e38f, asum = 0.0f;
  #pragma unroll
  for (int j = 0; j < 8; ++j) {
    const int t = tid + 256 * j;
    const float sc = s_z[t];
    zreg[j] = sc;
    smax = fmaxf(smax, sc);
    asum += attn[rowA + t];
  }
  const float m  = block_reduce_f(smax, 0, s_ftmp, &s_fres, tid);
  const float te = block_reduce_f(asum, 1, s_ftmp, &s_fres, tid);
  const float Kf = fmaxf(1.0f, rintf(RHO * te));   // round-half-even like jnp.round

  float esum = 0.0f;
  #pragma unroll
  for (int j = 0; j < 8; ++j) {
    zreg[j] = expf(zreg[j] - m);
    esum += zreg[j];
  }
  const float S = block_reduce_f(esum, 1, s_ftmp, &s_fres, tid);

  #pragma unroll
  for (int j = 0; j < 8; ++j) {
    const int t = tid + 256 * j;
    const float z = Kf * (zreg[j] / S);
    zreg[j] = z;
    s_z[t]  = z;
    out[(size_t)B_ * T_ + rowA + t] = z;   // second output: z
  }
  __syncthreads();

  // ---------------- Phase 3: exact top-K (stable ties by index) ----------------
  // z >= 0 so IEEE bit patterns are monotone: binary-search the K-th largest.
  const int Kc = (int)Kf;
  unsigned lo = 0u, hi = 0x7f800000u;           // cnt(0)=T >= K, cnt(inf)=0 < K
  while (hi - lo > 1u) {
    const unsigned mid = lo + ((hi - lo) >> 1);
    int c = 0;
    #pragma unroll
    for (int j = 0; j < 8; ++j) c += (__float_as_uint(zreg[j]) >= mid) ? 1 : 0;
    const int tot = block_reduce_i(c, s_itmp, &s_ires, tid);
    if (tot >= Kc) lo = mid; else hi = mid;
  }
  int cg = 0;
  #pragma unroll
  for (int j = 0; j < 8; ++j) cg += (__float_as_uint(zreg[j]) > lo) ? 1 : 0;
  const int c_gt = block_reduce_i(cg, s_itmp, &s_ires, tid);
  const int R = Kc - c_gt;   // #ties at theta accepted, lowest index first (>=1)

  #pragma unroll
  for (int j = 0; j < 8; ++j) {
    const int t = tid + 256 * j;
    const unsigned u = __float_as_uint(zreg[j]);
    float h = 0.0f;
    if (u > lo) {
      h = 1.0f;
    } else if (u == lo) {
      int eq = 0;
      for (int tp = 0; tp < t; ++tp) eq += (__float_as_uint(s_z[tp]) == lo) ? 1 : 0;
      if (eq < R) h = 1.0f;
    }
    out[rowA + t] = h;   // first output: g == h in forward
  }
}

extern "C" void kernel_launch(void* const* d_in, const int* in_sizes, int n_in,
                              void* d_out, int out_size, void* d_ws, size_t ws_size,
                              hipStream_t stream) {
  (void)in_sizes; (void)n_in; (void)out_size; (void)d_ws; (void)ws_size;
  const float* emb  = (const float*)d_in[0];
  const float* attn = (const float*)d_in[1];
  const float* W    = (const float*)d_in[2];
  const float* bptr = (const float*)d_in[3];
  float* out = (float*)d_out;
  hipLaunchKernelGGL(rationale_selector_kernel, dim3(B_), dim3(256), 0, stream,
                     emb, attn, W, bptr, out);
}